// Building_block_55121610277266
// MI455X (gfx1250) — compile-verified
//
#include <hip/hip_runtime.h>
#include <hip/hip_bf16.h>

#define B_    4
#define N_    8192
#define K_    16
#define C_    32
#define NK_   (N_*K_)

typedef float v2f __attribute__((ext_vector_type(2)));
typedef float v8f __attribute__((ext_vector_type(8)));

__device__ __forceinline__ v8f wmma4(v2f a, v2f b, v8f c) {
  // V_WMMA_F32_16X16X4_F32 : D(16x16,f32) = A(16x4,f32) x B(4x16,f32) + C
  return __builtin_amdgcn_wmma_f32_16x16x4_f32(false, a, false, b, (short)0, c, false, false);
}

__device__ __forceinline__ float red_max16(float v) {
  v = fmaxf(v, __shfl_xor(v, 1));
  v = fmaxf(v, __shfl_xor(v, 2));
  v = fmaxf(v, __shfl_xor(v, 4));
  v = fmaxf(v, __shfl_xor(v, 8));
  return v;
}
__device__ __forceinline__ float red_sum16(float v) {
  v += __shfl_xor(v, 1);
  v += __shfl_xor(v, 2);
  v += __shfl_xor(v, 4);
  v += __shfl_xor(v, 8);
  return v;
}

// ---------------------------------------------------------------------------
// Stage: relative position encoding + mlp1 (10 -> 32), pre-BN output
// ---------------------------------------------------------------------------
__global__ __launch_bounds__(256) void relpos_mlp1_kernel(
    const float* __restrict__ xyz, const int* __restrict__ nidx,
    const float* __restrict__ w, const float* __restrict__ bias,
    float* __restrict__ out /* [B,32,N,K] */)
{
  __shared__ float sW[320];
  __shared__ float sB[32];
  for (int i = threadIdx.x; i < 320; i += 256) sW[i] = w[i];
  if (threadIdx.x < 32) sB[threadIdx.x] = bias[threadIdx.x];
  __syncthreads();

  int e = blockIdx.x * 256 + threadIdx.x;          // e = (b*N+n)*K + k
  int k = e & 15;
  int n = (e >> 4) & (N_ - 1);
  int b = e >> 17;
  int idx = nidx[e];
  const float* pb = xyz + (size_t)b * N_ * 3;
  float tx = pb[n*3+0], ty = pb[n*3+1], tz = pb[n*3+2];
  float nx = pb[idx*3+0], ny = pb[idx*3+1], nz = pb[idx*3+2];
  float rx = tx-nx, ry = ty-ny, rz = tz-nz;
  float f[10] = { sqrtf(rx*rx+ry*ry+rz*rz), rx, ry, rz, tx, ty, tz, nx, ny, nz };
  #pragma unroll
  for (int c = 0; c < 32; ++c) {
    float acc = sB[c];
    #pragma unroll
    for (int i = 0; i < 10; ++i) acc += sW[c*10+i] * f[i];
    out[(b*32 + c)*NK_ + n*K_ + k] = acc;
  }
}

// ---------------------------------------------------------------------------
// BatchNorm (training mode) : per-channel sum/sumsq, then apply + LeakyReLU
// ---------------------------------------------------------------------------
__global__ __launch_bounds__(256) void bn_stats_kernel(
    const float* __restrict__ src, int Cch, int S,
    float* __restrict__ sums, float* __restrict__ sumsq)
{
  int c = blockIdx.x;
  float a = 0.f, a2 = 0.f;
  for (int b = 0; b < B_; ++b) {
    const float* p = src + (size_t)(b * Cch + c) * S;
    for (int i = threadIdx.x; i < S; i += 256) { float v = p[i]; a += v; a2 += v*v; }
  }
  __shared__ float r1[256], r2[256];
  r1[threadIdx.x] = a; r2[threadIdx.x] = a2;
  __syncthreads();
  for (int s = 128; s > 0; s >>= 1) {
    if (threadIdx.x < s) { r1[threadIdx.x] += r1[threadIdx.x+s]; r2[threadIdx.x] += r2[threadIdx.x+s]; }
    __syncthreads();
  }
  if (threadIdx.x == 0) { sums[c] = r1[0]; sumsq[c] = r2[0]; }
}

__global__ __launch_bounds__(256) void bn_apply_kernel(
    const float* __restrict__ src, float* __restrict__ dst,
    const float* __restrict__ sums, const float* __restrict__ sumsq,
    const float* __restrict__ g, const float* __restrict__ bt, int Cch, int S)
{
  long e = (long)blockIdx.x * 256 + threadIdx.x;   // over [B,Cch,S]
  int c = (int)((e / S) % Cch);
  float cnt = (float)B_ * (float)S;
  float mean = sums[c] / cnt;
  float var  = sumsq[c] / cnt - mean * mean;
  float y = (src[e] - mean) * rsqrtf(var + 1e-5f) * g[c] + bt[c];
  dst[e] = (y >= 0.f) ? y : 0.2f * y;
}

// ---------------------------------------------------------------------------
// Neighbor gather:  dst[b,c,n,k] = src[b,c, nidx[b,n,k]]   (src = [B,32,N])
// ---------------------------------------------------------------------------
__global__ __launch_bounds__(256) void gather_kernel(
    const float* __restrict__ src, const int* __restrict__ nidx, float* __restrict__ dst)
{
  int e = blockIdx.x * 256 + threadIdx.x;          // e = ((b*32+c)*N+n)*K+k
  int k = e & 15;
  int n = (e >> 4) & (N_ - 1);
  int c = (e >> 17) & 31;
  int b = e >> 22;
  int idx = nidx[((b << 13) + n) * K_ + k];
  dst[e] = src[((b << 5) + c) * N_ + idx];
}

// ---------------------------------------------------------------------------
// Squeeze-Excite: per-(b,c) mean, tiny MLP, scale apply
// ---------------------------------------------------------------------------
__global__ __launch_bounds__(256) void se_mean_kernel(const float* __restrict__ src, float* __restrict__ smean)
{
  int bc = blockIdx.x;                             // b*32+c
  const float* p = src + (size_t)bc * NK_;
  float a = 0.f;
  for (int i = threadIdx.x; i < NK_; i += 256) a += p[i];
  __shared__ float r[256];
  r[threadIdx.x] = a; __syncthreads();
  for (int s = 128; s > 0; s >>= 1) { if (threadIdx.x < s) r[threadIdx.x] += r[threadIdx.x+s]; __syncthreads(); }
  if (threadIdx.x == 0) smean[bc] = r[0] * (1.f / (float)NK_);
}

__global__ void se_fc_kernel(const float* __restrict__ smean,
                             const float* __restrict__ w1, const float* __restrict__ b1,
                             const float* __restrict__ w2, const float* __restrict__ b2,
                             float* __restrict__ scale)
{
  int t = threadIdx.x;                             // 128 threads = (b,c)
  int b = t >> 5, c = t & 31;
  float t1[8];
  #pragma unroll
  for (int j = 0; j < 8; ++j) {
    float z = b1[j];
    #pragma unroll
    for (int cc = 0; cc < 32; ++cc) z += w1[j*32+cc] * smean[b*32+cc];
    t1[j] = z / (1.f + expf(-z));                  // silu
  }
  float z = b2[c];
  #pragma unroll
  for (int j = 0; j < 8; ++j) z += w2[c*8+j] * t1[j];
  scale[t] = 1.f / (1.f + expf(-z));               // sigmoid
}

__global__ __launch_bounds__(256) void se_apply_kernel(float* __restrict__ buf, const float* __restrict__ scale)
{
  int e = blockIdx.x * 256 + threadIdx.x;
  buf[e] *= scale[e >> 17];                        // bc = e / NK_
}

// ---------------------------------------------------------------------------
// Multi-head attention over each K=16 neighborhood (WMMA f32 16x16x4).
// One wave per token group. Output written with the reference's raw reshape:
// flat ((b*N+n)*K + k)*C + c, later read as [B,C,N,K].
// ---------------------------------------------------------------------------
__global__ __launch_bounds__(128) void attn_kernel(
    const float* __restrict__ fn,                  // [B,32,N,K]
    const float* __restrict__ w_in, const float* __restrict__ b_in,    // [96,32],[96]
    const float* __restrict__ w_out, const float* __restrict__ b_out,  // [32,32],[32]
    float* __restrict__ out)
{
  const int lane = threadIdx.x & 31;
  const int wv   = threadIdx.x >> 5;               // 0..3
  __shared__ float sWin[96*32];
  __shared__ float sBin[96];
  __shared__ float sWout[32*32];
  __shared__ float sBout[32];
  __shared__ float sY[4][16*32];
  __shared__ float sQKV[4][16*96];
  __shared__ float sP[4][16*16];
  __shared__ float sO[4][16*32];

  for (int i = threadIdx.x; i < 96*32; i += 128) sWin[i] = w_in[i];
  for (int i = threadIdx.x; i < 96;    i += 128) sBin[i] = b_in[i];
  for (int i = threadIdx.x; i < 32*32; i += 128) sWout[i] = w_out[i];
  if (threadIdx.x < 32) sBout[threadIdx.x] = b_out[threadIdx.x];
  __syncthreads();

  const int m = blockIdx.x * 4 + wv;               // token group in [0, B*N)
  const int b = m >> 13, n = m & (N_ - 1);
  float* Y   = sY[wv];
  float* QKV = sQKV[wv];
  float* P   = sP[wv];
  float* O   = sO[wv];

  // stage y[k][c]
  for (int i = lane; i < 16*32; i += 32) {
    int k = i >> 5, c = i & 31;
    Y[i] = fn[((b*32 + c)*N_ + n)*K_ + k];
  }

  const int mrow = lane & 15;
  const int half = (lane >= 16) ? 8 : 0;
  const int ksel = (lane >= 16) ? 2 : 0;

  // qkv = y(16x32) @ Win^T(32x96)
  for (int t = 0; t < 6; ++t) {
    v8f acc = {};
    #pragma unroll
    for (int kk = 0; kk < 8; ++kk) {
      int k0 = kk*4 + ksel;
      v2f a, bb;
      a.x  = Y[mrow*32 + k0];       a.y  = Y[mrow*32 + k0 + 1];
      int col = t*16 + mrow;
      bb.x = sWin[col*32 + k0];     bb.y = sWin[col*32 + k0 + 1];
      acc = wmma4(a, bb, acc);
    }
    int col = t*16 + mrow;
    float bias = sBin[col];
    #pragma unroll
    for (int j = 0; j < 8; ++j) QKV[(j + half)*96 + col] = acc[j] + bias;
  }

  const float iscale = 0.3535533905932738f;        // 1/sqrt(8)
  for (int h = 0; h < 4; ++h) {
    // scores S = Q_h(16x8) @ K_h^T
    v8f s = {};
    #pragma unroll
    for (int kk = 0; kk < 2; ++kk) {
      int d0 = h*8 + kk*4 + ksel;
      v2f a, bb;
      a.x  = QKV[mrow*96 + d0];        a.y  = QKV[mrow*96 + d0 + 1];
      bb.x = QKV[mrow*96 + 32 + d0];   bb.y = QKV[mrow*96 + 32 + d0 + 1];
      s = wmma4(a, bb, s);
    }
    // softmax over the 16 columns (lanes within each half-wave)
    #pragma unroll
    for (int j = 0; j < 8; ++j) {
      float v  = s[j] * iscale;
      float mx = red_max16(v);
      float e  = expf(v - mx);
      float sm = red_sum16(e);
      s[j] = e / sm;
    }
    #pragma unroll
    for (int j = 0; j < 8; ++j) P[(j + half)*16 + mrow] = s[j];

    // o_h = P(16x16) @ V_h(16x8)  (B cols 8..15 padded with zero)
    v8f o = {};
    #pragma unroll
    for (int kk = 0; kk < 4; ++kk) {
      int k0 = kk*4 + ksel;
      v2f a, bb;
      a.x = P[mrow*16 + k0];  a.y = P[mrow*16 + k0 + 1];
      int dcol = mrow;
      bb.x = (dcol < 8) ? QKV[k0*96 + 64 + h*8 + dcol] : 0.f;
      bb.y = (dcol < 8) ? QKV[(k0+1)*96 + 64 + h*8 + dcol] : 0.f;
      o = wmma4(a, bb, o);
    }
    if (mrow < 8) {
      #pragma unroll
      for (int j = 0; j < 8; ++j) O[(j + half)*32 + h*8 + mrow] = o[j];
    }
  }

  // out-proj: r = O(16x32) @ Wout^T(32x32), raw-reshape store
  size_t base = (size_t)m * (K_ * C_);
  for (int t = 0; t < 2; ++t) {
    v8f acc = {};
    #pragma unroll
    for (int kk = 0; kk < 8; ++kk) {
      int k0 = kk*4 + ksel;
      v2f a, bb;
      a.x  = O[mrow*32 + k0];            a.y  = O[mrow*32 + k0 + 1];
      int c = t*16 + mrow;
      bb.x = sWout[c*32 + k0];           bb.y = sWout[c*32 + k0 + 1];
      acc = wmma4(a, bb, acc);
    }
    int c = t*16 + mrow;
    float bias = sBout[c];
    #pragma unroll
    for (int j = 0; j < 8; ++j) out[base + (size_t)(j + half)*C_ + c] = acc[j] + bias;
  }
}

// ---------------------------------------------------------------------------
// Attention pooling part A: sc = softmax_K( fc_w(96x96) @ X(96x16) );
// agg[o] = sum_k X[o][k] * sc[o][k]   -> agg [B,96,N]
// ---------------------------------------------------------------------------
__global__ __launch_bounds__(128) void attpool_a_kernel(
    const float* __restrict__ f_nb, const float* __restrict__ f_xyz, const float* __restrict__ f_tr,
    const float* __restrict__ fc_w, float* __restrict__ agg)
{
  const int lane = threadIdx.x & 31;
  const int wv   = threadIdx.x >> 5;
  __shared__ float sFC[96*96];
  __shared__ float sX[4][96*16];
  for (int i = threadIdx.x; i < 96*96; i += 128) sFC[i] = fc_w[i];
  __syncthreads();

  const int m = blockIdx.x * 4 + wv;               // b*N+n
  const int b = m >> 13, n = m & (N_ - 1);
  float* X = sX[wv];
  for (int i = lane; i < 96*16; i += 32) {
    int c = i >> 4, k = i & 15;
    const float* src = (c < 32) ? f_nb : ((c < 64) ? f_xyz : f_tr);
    X[i] = src[(((b*32) + (c & 31))*N_ + n)*K_ + k];
  }

  const int mrow = lane & 15;
  const int half = (lane >= 16) ? 8 : 0;
  const int ksel = (lane >= 16) ? 2 : 0;

  for (int t = 0; t < 6; ++t) {
    v8f acc = {};
    for (int kk = 0; kk < 24; ++kk) {
      int k0 = kk*4 + ksel;
      v2f a, bb;
      a.x  = sFC[(t*16 + mrow)*96 + k0];  a.y  = sFC[(t*16 + mrow)*96 + k0 + 1];
      bb.x = X[k0*16 + mrow];             bb.y = X[(k0+1)*16 + mrow];
      acc = wmma4(a, bb, acc);
    }
    #pragma unroll
    for (int j = 0; j < 8; ++j) {
      int o = t*16 + j + half;
      float v  = acc[j];
      float mx = red_max16(v);
      float e  = expf(v - mx);
      float sm = red_sum16(e);
      float p  = e / sm;
      float r  = red_sum16(p * X[o*16 + mrow]);
      if (mrow == 0) agg[(b*96 + o)*N_ + n] = r;
    }
  }
}

// ---------------------------------------------------------------------------
// Attention pooling part B: prebn[b,o,n] = mb[o] + sum_c mw[o][c]*agg[b,c,n]
// ---------------------------------------------------------------------------
__global__ __launch_bounds__(128) void attpool_b_kernel(
    const float* __restrict__ agg, const float* __restrict__ mw, const float* __restrict__ mb,
    float* __restrict__ prebn, int Cout)
{
  const int lane = threadIdx.x & 31;
  const int wv   = threadIdx.x >> 5;
  __shared__ float sW[64*96];
  __shared__ float sB[64];
  for (int i = threadIdx.x; i < Cout*96; i += 128) sW[i] = mw[i];
  if (threadIdx.x < Cout) sB[threadIdx.x] = mb[threadIdx.x];
  __syncthreads();

  const int wid = blockIdx.x * 4 + wv;             // B*(N/16) waves
  const int b  = wid / (N_/16);
  const int n0 = (wid % (N_/16)) * 16 + (lane & 15);
  const int ksel = (lane >= 16) ? 2 : 0;
  const int half = (lane >= 16) ? 8 : 0;

  v2f bf[24];
  #pragma unroll
  for (int kk = 0; kk < 24; ++kk) {
    int k0 = kk*4 + ksel;
    bf[kk].x = agg[(b*96 + k0)*N_ + n0];
    bf[kk].y = agg[(b*96 + k0 + 1)*N_ + n0];
  }
  for (int mt = 0; mt < Cout/16; ++mt) {
    v8f acc = {};
    #pragma unroll
    for (int kk = 0; kk < 24; ++kk) {
      int k0 = kk*4 + ksel;
      v2f a;
      a.x = sW[(mt*16 + (lane & 15))*96 + k0];
      a.y = sW[(mt*16 + (lane & 15))*96 + k0 + 1];
      acc = wmma4(a, bf[kk], acc);
    }
    #pragma unroll
    for (int j = 0; j < 8; ++j) {
      int o = mt*16 + j + half;
      prebn[(b*Cout + o)*N_ + n0] = acc[j] + sB[o];
    }
  }
}

// ---------------------------------------------------------------------------
// In-place 32->32 pointwise conv (mlp2) with WMMA, pre-BN (bias included)
// ---------------------------------------------------------------------------
__global__ __launch_bounds__(128) void conv32_wmma_kernel(
    float* __restrict__ buf, const float* __restrict__ w, const float* __restrict__ bias)
{
  const int lane = threadIdx.x & 31;
  const int wv   = threadIdx.x >> 5;
  __shared__ float sW[32*32];
  __shared__ float sB[32];
  for (int i = threadIdx.x; i < 1024; i += 128) sW[i] = w[i];
  if (threadIdx.x < 32) sB[threadIdx.x] = bias[threadIdx.x];
  __syncthreads();

  const int wid = blockIdx.x * 4 + wv;             // B*NK/16 waves
  const int b  = wid / (NK_/16);
  const int s0 = (wid % (NK_/16)) * 16 + (lane & 15);
  const int ksel = (lane >= 16) ? 2 : 0;
  const int half = (lane >= 16) ? 8 : 0;
  const int mrow = lane & 15;

  v2f bf[8];
  #pragma unroll
  for (int kk = 0; kk < 8; ++kk) {
    int k0 = kk*4 + ksel;
    bf[kk].x = buf[(b*32 + k0)*NK_ + s0];
    bf[kk].y = buf[(b*32 + k0 + 1)*NK_ + s0];
  }
  v8f acc0 = {}, acc1 = {};
  #pragma unroll
  for (int kk = 0; kk < 8; ++kk) {
    int k0 = kk*4 + ksel;
    v2f a0, a1;
    a0.x = sW[mrow*32 + k0];        a0.y = sW[mrow*32 + k0 + 1];
    a1.x = sW[(16 + mrow)*32 + k0]; a1.y = sW[(16 + mrow)*32 + k0 + 1];
    acc0 = wmma4(a0, bf[kk], acc0);
    acc1 = wmma4(a1, bf[kk], acc1);
  }
  #pragma unroll
  for (int j = 0; j < 8; ++j) {
    buf[(b*32 + j + half)*NK_ + s0]      = acc0[j] + sB[j + half];
    buf[(b*32 + 16 + j + half)*NK_ + s0] = acc1[j] + sB[16 + j + half];
  }
}

// ---------------------------------------------------------------------------
// Brute-force kNN (K=16) per batch; register-sorted top-16, LDS point tiles
// ---------------------------------------------------------------------------
__global__ __launch_bounds__(256) void knn_kernel(const float* __restrict__ xyz, int* __restrict__ kn)
{
  const int b = blockIdx.x / (N_/256);
  const int n = (blockIdx.x % (N_/256)) * 256 + threadIdx.x;
  const float* P = xyz + (size_t)b * N_ * 3;
  float qx = P[n*3+0], qy = P[n*3+1], qz = P[n*3+2];
  float bd[16]; int bi[16];
  #pragma unroll
  for (int i = 0; i < 16; ++i) { bd[i] = 1e30f; bi[i] = 0; }
  __shared__ float tile[256*3];
  for (int t0 = 0; t0 < N_; t0 += 256) {
    __syncthreads();
    for (int i = threadIdx.x; i < 256*3; i += 256) tile[i] = P[t0*3 + i];
    __syncthreads();
    if (t0 + 256 < N_) __builtin_prefetch(&P[(t0 + 256)*3 + threadIdx.x], 0, 0);
    for (int j = 0; j < 256; ++j) {
      float dx = qx - tile[j*3+0], dy = qy - tile[j*3+1], dz = qz - tile[j*3+2];
      float d = dx*dx + dy*dy + dz*dz;
      int pj = t0 + j;
      if (pj != n && d < bd[15]) {
        bd[15] = d; bi[15] = pj;
        #pragma unroll
        for (int q = 15; q > 0; --q) {
          if (bd[q] < bd[q-1]) {
            float td = bd[q]; bd[q] = bd[q-1]; bd[q-1] = td;
            int   ti = bi[q]; bi[q] = bi[q-1]; bi[q-1] = ti;
          }
        }
      }
    }
  }
  #pragma unroll
  for (int i = 0; i < 16; ++i) kn[((size_t)b*N_ + n)*K_ + i] = bi[i];
}

// ---------------------------------------------------------------------------
// GCN: two linear layers + neighbor aggregation ( (h + sum_nb h)/17 + b )
// ---------------------------------------------------------------------------
__global__ __launch_bounds__(256) void gcn_lin1_kernel(
    const float* __restrict__ f_agg /*[B,32,N]*/, const float* __restrict__ w /*[16,32]*/,
    float* __restrict__ h1 /*[B,N,16]*/)
{
  __shared__ float sW[512];
  for (int i = threadIdx.x; i < 512; i += 256) sW[i] = w[i];
  __syncthreads();
  int bn = blockIdx.x * 256 + threadIdx.x;
  int b = bn >> 13, n = bn & (N_ - 1);
  float x[32];
  #pragma unroll
  for (int c = 0; c < 32; ++c) x[c] = f_agg[(b*32 + c)*N_ + n];
  #pragma unroll
  for (int j = 0; j < 16; ++j) {
    float acc = 0.f;
    #pragma unroll
    for (int c = 0; c < 32; ++c) acc += sW[j*32+c] * x[c];
    h1[bn*16 + j] = acc;
  }
}

__global__ __launch_bounds__(256) void gcn_agg1_kernel(
    const float* __restrict__ h1, const int* __restrict__ kn,
    const float* __restrict__ bias, float* __restrict__ y1)
{
  int e = blockIdx.x * 256 + threadIdx.x;          // (b*N+n)*16 + j
  int j = e & 15, bn = e >> 4, b = bn >> 13;
  const int* kb = kn + (size_t)bn * K_;
  float acc = h1[bn*16 + j];
  for (int k = 0; k < 16; ++k) acc += h1[((b << 13) + kb[k])*16 + j];
  float v = acc * (1.f/17.f) + bias[j];
  y1[e] = fmaxf(0.f, v);
}

__global__ __launch_bounds__(256) void gcn_lin2_kernel(
    const float* __restrict__ y1 /*[B,N,16]*/, const float* __restrict__ w /*[32,16]*/,
    float* __restrict__ h2 /*[B,N,32]*/)
{
  __shared__ float sW[512];
  for (int i = threadIdx.x; i < 512; i += 256) sW[i] = w[i];
  __syncthreads();
  int bn = blockIdx.x * 256 + threadIdx.x;
  float x[16];
  #pragma unroll
  for (int j = 0; j < 16; ++j) x[j] = y1[bn*16 + j];
  #pragma unroll
  for (int c = 0; c < 32; ++c) {
    float acc = 0.f;
    #pragma unroll
    for (int j = 0; j < 16; ++j) acc += sW[c*16+j] * x[j];
    h2[bn*32 + c] = acc;
  }
}

__global__ __launch_bounds__(256) void gcn_agg2_kernel(
    const float* __restrict__ h2, const int* __restrict__ kn,
    const float* __restrict__ bias, float* __restrict__ f_agg)
{
  int e = blockIdx.x * 256 + threadIdx.x;          // (b*N+n)*32 + c
  int c = e & 31, bn = e >> 5, b = bn >> 13, n = bn & (N_ - 1);
  const int* kb = kn + (size_t)bn * K_;
  float acc = h2[bn*32 + c];
  for (int k = 0; k < 16; ++k) acc += h2[((b << 13) + kb[k])*32 + c];
  f_agg[(b*32 + c)*N_ + n] += acc * (1.f/17.f) + bias[c];
}

// ---------------------------------------------------------------------------
extern "C" void kernel_launch(void* const* d_in, const int* in_sizes, int n_in,
                              void* d_out, int out_size, void* d_ws, size_t ws_size,
                              hipStream_t stream)
{
  (void)in_sizes; (void)n_in; (void)out_size; (void)ws_size;
  const float* xyz       = (const float*)d_in[0];
  const float* feature   = (const float*)d_in[1];
  const int*   nidx      = (const int*)  d_in[2];
  const float* mlp1_w    = (const float*)d_in[3];
  const float* mlp1_b    = (const float*)d_in[4];
  const float* bn1_g     = (const float*)d_in[5];
  const float* bn1_b     = (const float*)d_in[6];
  const float* se_w1     = (const float*)d_in[7];
  const float* se_b1     = (const float*)d_in[8];
  const float* se_w2     = (const float*)d_in[9];
  const float* se_b2     = (const float*)d_in[10];
  const float* attn_in_w = (const float*)d_in[11];
  const float* attn_in_b = (const float*)d_in[12];
  const float* attn_out_w= (const float*)d_in[13];
  const float* attn_out_b= (const float*)d_in[14];
  const float* ap1_fc_w  = (const float*)d_in[15];
  const float* ap1_mlp_w = (const float*)d_in[16];
  const float* ap1_mlp_b = (const float*)d_in[17];
  const float* ap1_bn_g  = (const float*)d_in[18];
  const float* ap1_bn_b  = (const float*)d_in[19];
  const float* g1_w      = (const float*)d_in[20];
  const float* g1_b      = (const float*)d_in[21];
  const float* g2_w      = (const float*)d_in[22];
  const float* g2_b      = (const float*)d_in[23];
  const float* mlp2_w    = (const float*)d_in[24];
  const float* mlp2_b    = (const float*)d_in[25];
  const float* bn2_g     = (const float*)d_in[26];
  const float* bn2_b     = (const float*)d_in[27];
  const float* ap2_fc_w  = (const float*)d_in[28];
  const float* ap2_mlp_w = (const float*)d_in[29];
  const float* ap2_mlp_b = (const float*)d_in[30];
  const float* ap2_bn_g  = (const float*)d_in[31];
  const float* ap2_bn_b  = (const float*)d_in[32];

  float* ws     = (float*)d_ws;
  float* f_xyz  = ws;                    // 16,777,216
  float* f_nb   = f_xyz + 16777216;      // 16,777,216
  float* f_tr   = f_nb  + 16777216;      // 16,777,216
  float* aggb   = f_tr  + 16777216;      // 3,145,728  [B,96,N]
  float* prebn  = aggb  + 3145728;       // 2,097,152  [B,<=64,N]
  float* f_agg  = prebn + 2097152;       // 1,048,576  [B,32,N]
  float* h1     = f_agg + 1048576;       // 524,288
  float* y1     = h1    + 524288;        // 524,288
  float* h2     = y1    + 524288;        // 1,048,576
  float* sums   = h2    + 1048576;       // 128
  float* sumsq  = sums  + 128;           // 128
  float* semean = sumsq + 128;           // 128
  float* sescal = semean+ 128;           // 128
  int*   knbuf  = (int*)(sescal + 128);  // 524,288 ints

  const int GBIG = (B_*32*NK_) / 256;    // 65536 (elementwise over [B,32,N,K])

  // ---------------- Stage 1 ----------------
  relpos_mlp1_kernel<<<(B_*NK_)/256, 256, 0, stream>>>(xyz, nidx, mlp1_w, mlp1_b, f_xyz);
  bn_stats_kernel<<<32, 256, 0, stream>>>(f_xyz, 32, NK_, sums, sumsq);
  bn_apply_kernel<<<GBIG, 256, 0, stream>>>(f_xyz, f_xyz, sums, sumsq, bn1_g, bn1_b, 32, NK_);

  gather_kernel<<<GBIG, 256, 0, stream>>>(feature, nidx, f_nb);
  se_mean_kernel<<<128, 256, 0, stream>>>(f_nb, semean);
  se_fc_kernel<<<1, 128, 0, stream>>>(semean, se_w1, se_b1, se_w2, se_b2, sescal);
  se_apply_kernel<<<GBIG, 256, 0, stream>>>(f_nb, sescal);

  attn_kernel<<<(B_*N_)/4, 128, 0, stream>>>(f_nb, attn_in_w, attn_in_b, attn_out_w, attn_out_b, f_tr);

  attpool_a_kernel<<<(B_*N_)/4, 128, 0, stream>>>(f_nb, f_xyz, f_tr, ap1_fc_w, aggb);
  attpool_b_kernel<<<(B_*(N_/16))/4, 128, 0, stream>>>(aggb, ap1_mlp_w, ap1_mlp_b, prebn, 32);
  bn_stats_kernel<<<32, 256, 0, stream>>>(prebn, 32, N_, sums, sumsq);
  bn_apply_kernel<<<(B_*32*N_)/256, 256, 0, stream>>>(prebn, f_agg, sums, sumsq, ap1_bn_g, ap1_bn_b, 32, N_);

  knn_kernel<<<B_*(N_/256), 256, 0, stream>>>(xyz, knbuf);
  gcn_lin1_kernel<<<(B_*N_)/256, 256, 0, stream>>>(f_agg, g1_w, h1);
  gcn_agg1_kernel<<<(B_*N_*16)/256, 256, 0, stream>>>(h1, knbuf, g1_b, y1);
  gcn_lin2_kernel<<<(B_*N_)/256, 256, 0, stream>>>(y1, g2_w, h2);
  gcn_agg2_kernel<<<(B_*N_*32)/256, 256, 0, stream>>>(h2, knbuf, g2_b, f_agg);

  // ---------------- Stage 2 ----------------
  conv32_wmma_kernel<<<(B_*(NK_/16))/4, 128, 0, stream>>>(f_xyz, mlp2_w, mlp2_b);
  bn_stats_kernel<<<32, 256, 0, stream>>>(f_xyz, 32, NK_, sums, sumsq);
  bn_apply_kernel<<<GBIG, 256, 0, stream>>>(f_xyz, f_xyz, sums, sumsq, bn2_g, bn2_b, 32, NK_);

  gather_kernel<<<GBIG, 256, 0, stream>>>(f_agg, nidx, f_nb);
  se_mean_kernel<<<128, 256, 0, stream>>>(f_nb, semean);
  se_fc_kernel<<<1, 128, 0, stream>>>(semean, se_w1, se_b1, se_w2, se_b2, sescal);
  se_apply_kernel<<<GBIG, 256, 0, stream>>>(f_nb, sescal);

  attn_kernel<<<(B_*N_)/4, 128, 0, stream>>>(f_nb, attn_in_w, attn_in_b, attn_out_w, attn_out_b, f_tr);

  attpool_a_kernel<<<(B_*N_)/4, 128, 0, stream>>>(f_nb, f_xyz, f_tr, ap2_fc_w, aggb);
  attpool_b_kernel<<<(B_*(N_/16))/4, 128, 0, stream>>>(aggb, ap2_mlp_w, ap2_mlp_b, prebn, 64);
  bn_stats_kernel<<<64, 256, 0, stream>>>(prebn, 64, N_, sums, sumsq);
  bn_apply_kernel<<<(B_*64*N_)/256, 256, 0, stream>>>(prebn, (float*)d_out, sums, sumsq, ap2_bn_g, ap2_bn_b, 64, N_);
}